// TorchSMPL4Garment_8074538517083
// MI455X (gfx1250) — compile-verified
//
#include <hip/hip_runtime.h>
#include <math.h>

#define NJ     24
#define NBATCH 256
#define NV     27554
#define NE     (NV * 3)     // 82662 elements per batch
#define CSW    12           // shape coeff row width (10 + 2 zero pad)
#define CPW    208          // pose coeff row width (207 + 1 zero pad)

typedef __attribute__((ext_vector_type(2))) float v2f;
typedef __attribute__((ext_vector_type(8))) float v8f;

__constant__ int c_parents[NJ] = {0,0,0,0,1,2,3,4,5,6,7,8,9,9,9,12,13,14,16,17,18,19,20,21};

// ---------------------------------------------------------------------------
// Kernel A: SJ[k,j,c] = sum_v shapedirs[k,3v+c] * Jr[v,j]   (k<10)
//           Jbase[j,c] = sum_v v_template[v,c]  * Jr[v,j]   (k==10)
// ---------------------------------------------------------------------------
__global__ void k_joint_reduce(const float* __restrict__ vt,
                               const float* __restrict__ sd,
                               const float* __restrict__ jr,
                               float* __restrict__ SJ,
                               float* __restrict__ Jb) {
    const int k = blockIdx.x / NJ;   // 0..10
    const int j = blockIdx.x % NJ;
    const int t = threadIdx.x;
    float s0 = 0.f, s1 = 0.f, s2 = 0.f;
    if (k < 10) {
        const float* row = sd + (size_t)k * NE;
        for (int v = t; v < NV; v += 256) {
            const float w = jr[v * NJ + j];
            s0 += row[3*v+0] * w; s1 += row[3*v+1] * w; s2 += row[3*v+2] * w;
        }
    } else {
        for (int v = t; v < NV; v += 256) {
            const float w = jr[v * NJ + j];
            s0 += vt[3*v+0] * w; s1 += vt[3*v+1] * w; s2 += vt[3*v+2] * w;
        }
    }
    __shared__ float red[256][3];
    red[t][0] = s0; red[t][1] = s1; red[t][2] = s2;
    __syncthreads();
    for (int off = 128; off > 0; off >>= 1) {
        if (t < off) {
            red[t][0] += red[t+off][0];
            red[t][1] += red[t+off][1];
            red[t][2] += red[t+off][2];
        }
        __syncthreads();
    }
    if (t == 0) {
        float* dst = (k < 10) ? (SJ + (k * NJ + j) * 3) : (Jb + j * 3);
        dst[0] = red[0][0]; dst[1] = red[0][1]; dst[2] = red[0][2];
    }
}

// ---------------------------------------------------------------------------
// Kernel B: per batch (one wave): Rodrigues -> Rs, shape/pose coeff rows,
// J from SJ/Jbase, joint chain -> A[24][12]
// ---------------------------------------------------------------------------
__global__ void k_batch_prep(const float* __restrict__ theta,
                             const float* __restrict__ beta,
                             const float* __restrict__ SJ,
                             const float* __restrict__ Jb,
                             float* __restrict__ cshape,
                             float* __restrict__ cpose,
                             float* __restrict__ Afin) {
    const int b = blockIdx.x;
    const int t = threadIdx.x;   // 32 threads
    __shared__ float sR[NJ][9];
    __shared__ float sJ[NJ][3];
    __shared__ float sRg[NJ][9];
    __shared__ float sTg[NJ][3];

    if (t < NJ) {
        const float tx = theta[b*72 + 3*t + 0];
        const float ty = theta[b*72 + 3*t + 1];
        const float tz = theta[b*72 + 3*t + 2];
        const float ux = tx + 1e-8f, uy = ty + 1e-8f, uz = tz + 1e-8f;
        const float ang = sqrtf(ux*ux + uy*uy + uz*uz);
        const float inv = 1.0f / ang;
        const float h   = 0.5f * ang;
        float qw = cosf(h);
        const float s = sinf(h);
        float qx = s * tx * inv, qy = s * ty * inv, qz = s * tz * inv;
        const float qn = rsqrtf(qw*qw + qx*qx + qy*qy + qz*qz);
        qw *= qn; qx *= qn; qy *= qn; qz *= qn;
        sR[t][0] = qw*qw + qx*qx - qy*qy - qz*qz;
        sR[t][1] = 2.f*(qx*qy - qw*qz);
        sR[t][2] = 2.f*(qw*qy + qx*qz);
        sR[t][3] = 2.f*(qw*qz + qx*qy);
        sR[t][4] = qw*qw - qx*qx + qy*qy - qz*qz;
        sR[t][5] = 2.f*(qy*qz - qw*qx);
        sR[t][6] = 2.f*(qx*qz - qw*qy);
        sR[t][7] = 2.f*(qw*qx + qy*qz);
        sR[t][8] = qw*qw - qx*qx - qy*qy + qz*qz;
        float j0 = Jb[t*3+0], j1 = Jb[t*3+1], j2 = Jb[t*3+2];
        #pragma unroll
        for (int k = 0; k < 10; ++k) {
            const float bk = beta[b*10 + k];
            j0 += bk * SJ[(k*NJ + t)*3 + 0];
            j1 += bk * SJ[(k*NJ + t)*3 + 1];
            j2 += bk * SJ[(k*NJ + t)*3 + 2];
        }
        sJ[t][0] = j0; sJ[t][1] = j1; sJ[t][2] = j2;
    }
    __syncthreads();

    // shape coeff row: [beta(10) | 0 0]
    if (t < CSW) cshape[b*CSW + t] = (t < 10) ? beta[b*10 + t] : 0.f;
    // pose coeff row: [Rs[1:]-I (207) | 0]
    for (int k = t; k < CPW; k += 32) {
        float c = 0.f;
        if (k < 207) {
            const int ji = k / 9 + 1, rc = k % 9;
            c = sR[ji][rc] - ((rc == 0 || rc == 4 || rc == 8) ? 1.f : 0.f);
        }
        cpose[b*CPW + k] = c;
    }

    if (t == 0) {  // sequential joint chain (parents precede children)
        for (int q = 0; q < 9; ++q) sRg[0][q] = sR[0][q];
        sTg[0][0] = sJ[0][0]; sTg[0][1] = sJ[0][1]; sTg[0][2] = sJ[0][2];
        for (int i = 1; i < NJ; ++i) {
            const int p = c_parents[i];
            const float t0 = sJ[i][0]-sJ[p][0], t1 = sJ[i][1]-sJ[p][1], t2 = sJ[i][2]-sJ[p][2];
            for (int r = 0; r < 3; ++r) {
                const float p0 = sRg[p][r*3+0], p1 = sRg[p][r*3+1], p2 = sRg[p][r*3+2];
                sRg[i][r*3+0] = p0*sR[i][0] + p1*sR[i][3] + p2*sR[i][6];
                sRg[i][r*3+1] = p0*sR[i][1] + p1*sR[i][4] + p2*sR[i][7];
                sRg[i][r*3+2] = p0*sR[i][2] + p1*sR[i][5] + p2*sR[i][8];
                sTg[i][r]     = p0*t0 + p1*t1 + p2*t2 + sTg[p][r];
            }
        }
    }
    __syncthreads();

    if (t < NJ) {  // A[b,t] = [Rg | Tg - Rg*J]
        const float jx = sJ[t][0], jy = sJ[t][1], jz = sJ[t][2];
        for (int r = 0; r < 3; ++r) {
            const float r0 = sRg[t][r*3+0], r1 = sRg[t][r*3+1], r2 = sRg[t][r*3+2];
            const float tr = sTg[t][r] - (r0*jx + r1*jy + r2*jz);
            float* d = Afin + ((size_t)(b*NJ + t))*12 + r*4;
            d[0] = r0; d[1] = r1; d[2] = r2; d[3] = tr;
        }
    }
}

// ---------------------------------------------------------------------------
// Main fused kernel: one wave per (16-vertex x 16-batch) tile.
// Branch-free WMMA f32 16x16x4 K-loops (shape K=12, pose K=208), LDS staging,
// register-weight VALU skinning, direct output store.
// ---------------------------------------------------------------------------
#define WMMA4(ACC, A, B) \
    ACC = __builtin_amdgcn_wmma_f32_16x16x4_f32(false, (A), false, (B), (short)0, ACC, false, false)

__global__ void __launch_bounds__(32)
k_lbs_main(const float* __restrict__ vtemp,
           const float* __restrict__ sd,
           const float* __restrict__ pd,
           const float* __restrict__ wts,
           const float* __restrict__ cshape,
           const float* __restrict__ cpose,
           const float* __restrict__ Afin,
           float* __restrict__ out) {
    const int lane = threadIdx.x;
    const int v0 = blockIdx.x * 16;
    const int b0 = blockIdx.y * 16;
    const int e0 = v0 * 3;

    __shared__ __align__(16) float sA[16 * NJ * 12];  // A tile: 16 batches
    __shared__ float sP[16 * 52];                     // v_posed tile, padded stride

    for (int i = lane; i < 16 * NJ * 12; i += 32)
        sA[i] = Afin[(size_t)b0 * NJ * 12 + i];

    const int  c16 = lane & 15;
    const bool hi  = lane >= 16;
    const int  roff = hi ? 2 : 0;     // K-row offset for this lane half

    // Column indices, clamped once: out-of-range columns only feed vertices
    // that are never stored, so duplicated (finite) data is harmless.
    const int eA = min(e0 + c16,      NE - 1);
    const int eB = min(e0 + c16 + 16, NE - 1);
    const int eC = min(e0 + c16 + 32, NE - 1);

    v8f acc0 = {0,0,0,0,0,0,0,0};
    v8f acc1 = acc0, acc2 = acc0;

    // ---------------- shape segment: K rows 0..11 ----------------
    {
        const float* cs = cshape + (b0 + c16) * CSW + roff;
        const float* r  = sd + (size_t)roff * NE;
        #pragma unroll
        for (int kk = 0; kk < 2; ++kk) {           // rows 0..7: branch-free
            const v2f a = *(const v2f*)cs;
            v2f bf0, bf1, bf2;
            bf0.x = r[eA]; bf0.y = r[NE + eA];
            bf1.x = r[eB]; bf1.y = r[NE + eB];
            bf2.x = r[eC]; bf2.y = r[NE + eC];
            WMMA4(acc0, a, bf0); WMMA4(acc1, a, bf1); WMMA4(acc2, a, bf2);
            cs += 4; r += 4 * NE;
        }
        {   // rows 8..11: hi-lane coeff rows 10,11 are zero; clamp B rows to 9
            const v2f a = *(const v2f*)cs;
            const float* rA = sd + (size_t)(hi ? 9 : 8) * NE;
            const float* rB = sd + (size_t)9 * NE;
            v2f bf0, bf1, bf2;
            bf0.x = rA[eA]; bf0.y = rB[eA];
            bf1.x = rA[eB]; bf1.y = rB[eB];
            bf2.x = rA[eC]; bf2.y = rB[eC];
            WMMA4(acc0, a, bf0); WMMA4(acc1, a, bf1); WMMA4(acc2, a, bf2);
        }
    }
    // ---------------- pose segment: K rows 0..207 ----------------
    {
        const float* cp = cpose + (b0 + c16) * CPW + roff;
        const float* r  = pd + (size_t)roff * NE;
        #pragma unroll 3
        for (int kk = 0; kk < 51; ++kk) {          // rows 0..203: branch-free
            const v2f a = *(const v2f*)cp;
            v2f bf0, bf1, bf2;
            bf0.x = r[eA]; bf0.y = r[NE + eA];
            bf1.x = r[eB]; bf1.y = r[NE + eB];
            bf2.x = r[eC]; bf2.y = r[NE + eC];
            WMMA4(acc0, a, bf0); WMMA4(acc1, a, bf1); WMMA4(acc2, a, bf2);
            cp += 4; r += 4 * NE;
        }
        {   // rows 204..207: row 207 has zero coeff; clamp its B row to 206
            const v2f a = *(const v2f*)cp;
            const float* rA = pd + (size_t)(204 + roff) * NE;   // 204 / 206
            const float* rB = pd + (size_t)(hi ? 206 : 205) * NE;
            v2f bf0, bf1, bf2;
            bf0.x = rA[eA]; bf0.y = rB[eA];
            bf1.x = rA[eB]; bf1.y = rB[eB];
            bf2.x = rA[eC]; bf2.y = rB[eC];
            WMMA4(acc0, a, bf0); WMMA4(acc1, a, bf1); WMMA4(acc2, a, bf2);
        }
    }

    // Stage v_posed (+ v_template) to LDS. C layout: VGPR r -> rows r / r+8.
    {
        const int rbase = hi ? 8 : 0;
        const float tv0 = vtemp[eA];   // clamped cols: value unused for stores
        const float tv1 = vtemp[eB];
        const float tv2 = vtemp[eC];
        #pragma unroll
        for (int r = 0; r < 8; ++r) {
            sP[(r + rbase) * 52 + c16 +  0] = acc0[r] + tv0;
            sP[(r + rbase) * 52 + c16 + 16] = acc1[r] + tv1;
            sP[(r + rbase) * 52 + c16 + 32] = acc2[r] + tv2;
        }
    }
    __syncthreads();

    // Skinning: lane handles vertex c16 for 8 batches (b_local = 2i + hi).
    const int v = v0 + c16;
    const bool vok = v < NV;
    const int vc = vok ? v : (NV - 1);

    // Hoist the 24 weights into registers (6x float4, 16B-aligned rows).
    const float4* wq = (const float4*)(wts + (size_t)vc * NJ);
    const float4 W0 = wq[0], W1 = wq[1], W2 = wq[2], W3 = wq[3], W4 = wq[4], W5 = wq[5];
    const float wv[24] = {W0.x,W0.y,W0.z,W0.w, W1.x,W1.y,W1.z,W1.w,
                          W2.x,W2.y,W2.z,W2.w, W3.x,W3.y,W3.z,W3.w,
                          W4.x,W4.y,W4.z,W4.w, W5.x,W5.y,W5.z,W5.w};

    for (int i = 0; i < 8; ++i) {
        const int bl = 2 * i + (hi ? 1 : 0);
        const float x = sP[bl * 52 + 3 * c16 + 0];
        const float y = sP[bl * 52 + 3 * c16 + 1];
        const float z = sP[bl * 52 + 3 * c16 + 2];
        float M0=0,M1=0,M2=0,M3=0,M4=0,M5=0,M6=0,M7=0,M8=0,M9=0,M10=0,M11=0;
        const float* Ab = sA + bl * NJ * 12;
        #pragma unroll
        for (int j = 0; j < NJ; ++j) {
            const float w = wv[j];
            const float4 a0 = *(const float4*)(Ab + j * 12 + 0);
            const float4 a1 = *(const float4*)(Ab + j * 12 + 4);
            const float4 a2 = *(const float4*)(Ab + j * 12 + 8);
            M0 += w * a0.x;  M1 += w * a0.y;  M2  += w * a0.z;  M3  += w * a0.w;
            M4 += w * a1.x;  M5 += w * a1.y;  M6  += w * a1.z;  M7  += w * a1.w;
            M8 += w * a2.x;  M9 += w * a2.y;  M10 += w * a2.z;  M11 += w * a2.w;
        }
        if (vok) {
            float* o = out + ((size_t)(b0 + bl) * NV + v) * 3;
            o[0] = M0*x + M1*y + M2 *z + M3;
            o[1] = M4*x + M5*y + M6 *z + M7;
            o[2] = M8*x + M9*y + M10*z + M11;
        }
    }
}

// ---------------------------------------------------------------------------
extern "C" void kernel_launch(void* const* d_in, const int* in_sizes, int n_in,
                              void* d_out, int out_size, void* d_ws, size_t ws_size,
                              hipStream_t stream) {
    const float* theta = (const float*)d_in[0];
    const float* beta  = (const float*)d_in[1];
    const float* vt    = (const float*)d_in[2];
    const float* sd    = (const float*)d_in[3];
    const float* jr    = (const float*)d_in[4];
    const float* pd    = (const float*)d_in[5];
    const float* wts   = (const float*)d_in[6];
    float* out = (float*)d_out;

    // workspace (floats): SJ 720 | Jbase 72 (pad->800) | cshape 256*12 |
    //                     cpose 256*208 | A 256*288   (~523 KB total)
    float* ws     = (float*)d_ws;
    float* SJ     = ws;
    float* Jb     = ws + 720;
    float* cshape = ws + 800;
    float* cpose  = cshape + NBATCH * CSW;
    float* Afin   = cpose  + NBATCH * CPW;

    k_joint_reduce<<<dim3(11 * NJ), 256, 0, stream>>>(vt, sd, jr, SJ, Jb);
    k_batch_prep<<<dim3(NBATCH), 32, 0, stream>>>(theta, beta, SJ, Jb, cshape, cpose, Afin);
    dim3 grid((NV + 15) / 16, NBATCH / 16);
    k_lbs_main<<<grid, 32, 0, stream>>>(vt, sd, pd, wts, cshape, cpose, Afin, out);

    (void)in_sizes; (void)n_in; (void)out_size; (void)ws_size;
}